// SNN_BE_ResNet_35510789603332
// MI455X (gfx1250) — compile-verified
//
#include <hip/hip_runtime.h>
#include <math.h>

// Problem constants (match reference)
#define MM 4      // models
#define BB 128    // batch per model
#define CC 10     // classes
#define KK 32     // centroids
#define FF 512    // feature dim
#define HH 512    // hidden dim
// NZ = 512 (noise dim), proto rows per model = C*K = 320

typedef __attribute__((ext_vector_type(2))) float v2f;
typedef __attribute__((ext_vector_type(8))) float v8f;

// ---------------------------------------------------------------------------
// One 16x64 output strip of  D = (A ⊙ rs) @ B  using V_WMMA_F32_16X16X4_F32.
// A-frag (16x4 f32, ISA 7.12.2): lane l holds A[m=l&15, k = v + 2*(l>>4)],
// B-frag mirrored: lane l holds B[k = v + 2*(l>>4), n = l&15].
// The A fragment (and its BatchEnsemble r-scale) is loaded ONCE per k-step and
// reused across 4 adjacent column tiles -> 4 independent WMMAs per k-step.
// Arow = A + row0*lda ; Bcol = B + col0 ; rs = per-k input scale.
// ---------------------------------------------------------------------------
__device__ __forceinline__ void be_tile4_f32(const float* __restrict__ Arow, int lda,
                                             const float* __restrict__ rs,
                                             const float* __restrict__ Bcol, int ldb,
                                             int Kdim, int lane, v8f acc[4])
{
    const int mr = lane & 15;
    const int hi = lane >> 4;
    #pragma unroll 2
    for (int k = 0; k < Kdim; k += 4) {
        const int ka = k + 2 * hi;
        v2f a;
        a.x = Arow[mr * lda + ka]     * rs[ka];
        a.y = Arow[mr * lda + ka + 1] * rs[ka + 1];
        const float* bp = Bcol + ka * ldb + mr;
        #pragma unroll
        for (int t = 0; t < 4; ++t) {
            v2f b;
            b.x = bp[t * 16];
            b.y = bp[ldb + t * 16];
            acc[t] = __builtin_amdgcn_wmma_f32_16x16x4_f32(false, a, false, b,
                                                           (short)0, acc[t], false, false);
        }
    }
}

// ---------------------------------------------------------------------------
// Kernel 1: h = relu( ((noise ⊙ r1[:512]) @ W1[:512] + onehot-row) ⊙ s1 + b1 )
// per model: [320,512] x [512,512]; onehot contribution = r1[m,512+c]*W1[512+c,o]
// (c = row/32 is constant within any 16-row tile, so it hoists cleanly)
// ---------------------------------------------------------------------------
__global__ void k_proto_h(const float* __restrict__ noise,  // [320,512]
                          const float* __restrict__ W1,     // [522,512]
                          const float* __restrict__ r1,     // [4,522]
                          const float* __restrict__ s1,     // [4,512]
                          const float* __restrict__ b1,     // [4,512]
                          float* __restrict__ h)            // [4*320,512]
{
    const int lane = threadIdx.x & 31;
    const int wave = blockIdx.x * (blockDim.x >> 5) + (threadIdx.x >> 5);
    const int RT = 320 / 16, CTG = 512 / 64;                // 20 x 8 strip-tiles
    const int m    = wave / (RT * CTG);
    const int rem  = wave % (RT * CTG);
    const int row0 = (rem / CTG) * 16;
    const int col0 = (rem % CTG) * 64;

    v8f acc[4] = {};
    be_tile4_f32(noise + row0 * 512, 512, r1 + m * 522, W1 + col0, 512, 512, lane, acc);

    const int mr = lane & 15, hi = lane >> 4;
    const int c  = row0 >> 5;                  // class = row / K (constant per tile)
    const float rc = r1[m * 522 + 512 + c];
    #pragma unroll
    for (int t = 0; t < 4; ++t) {
        const int o = col0 + 16 * t + mr;
        const float add = rc * W1[(512 + c) * 512 + o];
        const float sv  = s1[m * 512 + o];
        const float bv  = b1[m * 512 + o];
        #pragma unroll
        for (int j = 0; j < 8; ++j) {
            const int p = row0 + j + 8 * hi;
            const float val = (acc[t][j] + add) * sv + bv;
            h[(m * 320 + p) * 512 + o] = fmaxf(val, 0.0f);
        }
    }
}

// ---------------------------------------------------------------------------
// Kernel 2: proto = ((h ⊙ r2) @ W2) ⊙ s2 + b2   per model [320,512]x[512,512]
// ---------------------------------------------------------------------------
__global__ void k_proto(const float* __restrict__ h,
                        const float* __restrict__ W2,      // [512,512]
                        const float* __restrict__ r2,      // [4,512]
                        const float* __restrict__ s2,      // [4,512]
                        const float* __restrict__ b2,      // [4,512]
                        float* __restrict__ proto)         // [4*320,512]
{
    const int lane = threadIdx.x & 31;
    const int wave = blockIdx.x * (blockDim.x >> 5) + (threadIdx.x >> 5);
    const int RT = 320 / 16, CTG = 512 / 64;
    const int m    = wave / (RT * CTG);
    const int rem  = wave % (RT * CTG);
    const int row0 = (rem / CTG) * 16;
    const int col0 = (rem % CTG) * 64;

    v8f acc[4] = {};
    be_tile4_f32(h + (m * 320 + row0) * 512, 512, r2 + m * 512,
                 W2 + col0, 512, 512, lane, acc);

    const int mr = lane & 15, hi = lane >> 4;
    #pragma unroll
    for (int t = 0; t < 4; ++t) {
        const int o = col0 + 16 * t + mr;
        const float sv = s2[m * 512 + o];
        const float bv = b2[m * 512 + o];
        #pragma unroll
        for (int j = 0; j < 8; ++j) {
            const int p = row0 + j + 8 * hi;
            proto[(m * 320 + p) * 512 + o] = acc[t][j] * sv + bv;
        }
    }
}

// ---------------------------------------------------------------------------
// Kernel 3: z = ((x ⊙ r_lw) @ W_lw) ⊙ s_lw   per model [128,512]x[512,5120]
// Dominant GEMM (~2.7 GFLOP): 16x64 strip per wave, A-frag reused 4x.
// ---------------------------------------------------------------------------
__global__ void k_z(const float* __restrict__ x,          // [4*128,512]
                    const float* __restrict__ W,          // [512,5120]
                    const float* __restrict__ r,          // [4,512]
                    const float* __restrict__ s,          // [4,5120]
                    float* __restrict__ z)                // [4*128,5120]
{
    const int lane = threadIdx.x & 31;
    const int wave = blockIdx.x * (blockDim.x >> 5) + (threadIdx.x >> 5);
    const int RT = 128 / 16, CTG = 5120 / 64;              // 8 x 80 strip-tiles
    const int m    = wave / (RT * CTG);
    const int rem  = wave % (RT * CTG);
    const int row0 = (rem / CTG) * 16;
    const int col0 = (rem % CTG) * 64;

    v8f acc[4] = {};
    be_tile4_f32(x + (m * 128 + row0) * 512, 512, r + m * 512,
                 W + col0, 5120, 512, lane, acc);

    const int mr = lane & 15, hi = lane >> 4;
    #pragma unroll
    for (int t = 0; t < 4; ++t) {
        const int o = col0 + 16 * t + mr;
        const float sv = s[m * 5120 + o];
        #pragma unroll
        for (int j = 0; j < 8; ++j) {
            const int p = row0 + j + 8 * hi;
            z[(m * 128 + p) * 5120 + o] = acc[t][j] * sv;
        }
    }
}

// ---------------------------------------------------------------------------
// Kernel 4: reduce proto over K:  psum[m,c,f] = Σ_k proto[m,c,k,f]
//                                 pnsum[m,c]  = Σ_{k,f} proto²
// one block per (m,c): 40 blocks x 256 threads
// ---------------------------------------------------------------------------
__global__ void k_psum(const float* __restrict__ proto,   // [4*320,512]
                       float* __restrict__ psum,          // [40,512]
                       float* __restrict__ pnsum)         // [40]
{
    const int mc = blockIdx.x;
    const int m  = mc / CC, c = mc % CC;
    __shared__ float red[256];
    float sq = 0.0f;
    for (int f = threadIdx.x; f < 512; f += 256) {
        float s = 0.0f;
        #pragma unroll 8
        for (int k = 0; k < KK; ++k) {
            const float v = proto[(m * 320 + c * KK + k) * 512 + f];
            s  += v;
            sq += v * v;
        }
        psum[mc * 512 + f] = s;
    }
    red[threadIdx.x] = sq;
    __syncthreads();
    for (int off = 128; off > 0; off >>= 1) {
        if ((int)threadIdx.x < off) red[threadIdx.x] += red[threadIdx.x + off];
        __syncthreads();
    }
    if (threadIdx.x == 0) pnsum[mc] = red[0];
}

// ---------------------------------------------------------------------------
// Kernel 5: out[mb,c] = exp( -(K·‖z_c‖² − 2·z_c·psum + pnsum) / (2·K·F·σ²) )
// (algebraic expansion of the [M,B,C,K,F] broadcast: Σ_{k,f}(z−p)² =
//  K‖z‖² − 2 z·Σ_k p + Σ_k‖p‖²) — one block per (m,b), wave per class.
// ---------------------------------------------------------------------------
__global__ void k_out(const float* __restrict__ z,        // [4*128,5120]
                      const float* __restrict__ psum,     // [40,512]
                      const float* __restrict__ pnsum,    // [40]
                      float* __restrict__ out)            // [4*128,10]
{
    const int mb   = blockIdx.x;
    const int m    = mb >> 7;                  // mb / 128
    const int c    = threadIdx.x >> 5;         // wave id = class
    const int lane = threadIdx.x & 31;
    const float* zr = z + (size_t)mb * 5120;
    const float* ps = psum + (m * CC + c) * 512;

    float zn = 0.0f, dp = 0.0f;
    for (int f = lane; f < 512; f += 32) {
        const float v = zr[f * CC + c];        // zt[m,b,c,f]
        zn += v * v;
        dp += v * ps[f];
    }
    #pragma unroll
    for (int off = 16; off > 0; off >>= 1) {
        zn += __shfl_xor(zn, off);
        dp += __shfl_xor(dp, off);
    }
    if (lane == 0) {
        // 1/(K*F*2*sigma^2) = 1/32768
        const float dist = -(32.0f * zn - 2.0f * dp + pnsum[m * CC + c]) * (1.0f / 32768.0f);
        out[mb * CC + c] = expf(dist);
    }
}

// ---------------------------------------------------------------------------
// Launch
// ---------------------------------------------------------------------------
extern "C" void kernel_launch(void* const* d_in, const int* in_sizes, int n_in,
                              void* d_out, int out_size, void* d_ws, size_t ws_size,
                              hipStream_t stream) {
    (void)in_sizes; (void)n_in; (void)out_size; (void)ws_size;
    const float* x     = (const float*)d_in[0];   // [512,512]
    const float* noise = (const float*)d_in[1];   // [320,512]
    const float* W_lw  = (const float*)d_in[2];   // [512,5120]
    const float* r_lw  = (const float*)d_in[3];   // [4,512]
    const float* s_lw  = (const float*)d_in[4];   // [4,5120]
    const float* W1    = (const float*)d_in[5];   // [522,512]
    const float* r1    = (const float*)d_in[6];   // [4,522]
    const float* s1    = (const float*)d_in[7];   // [4,512]
    const float* b1    = (const float*)d_in[8];   // [4,512]
    const float* W2    = (const float*)d_in[9];   // [512,512]
    const float* r2    = (const float*)d_in[10];  // [4,512]
    const float* s2    = (const float*)d_in[11];  // [4,512]
    const float* b2    = (const float*)d_in[12];  // [4,512]
    float* out = (float*)d_out;

    // workspace layout (floats): ~15.8 MB total
    float* ws       = (float*)d_ws;
    float* ws_h     = ws;                        // 4*320*512   = 655360
    float* ws_proto = ws_h     + 655360;         // 4*320*512   = 655360
    float* ws_z     = ws_proto + 655360;         // 4*128*5120  = 2621440
    float* ws_psum  = ws_z     + 2621440;        // 40*512      = 20480
    float* ws_pnsum = ws_psum  + 20480;          // 40

    // 8 waves / block, one 16x64 WMMA strip per wave
    k_proto_h<<<80,  256, 0, stream>>>(noise, W1, r1, s1, b1, ws_h);      // 640 strips
    k_proto  <<<80,  256, 0, stream>>>(ws_h, W2, r2, s2, b2, ws_proto);   // 640 strips
    k_z      <<<320, 256, 0, stream>>>(x, W_lw, r_lw, s_lw, ws_z);        // 2560 strips
    k_psum   <<<40,  256, 0, stream>>>(ws_proto, ws_psum, ws_pnsum);
    k_out    <<<512, 320, 0, stream>>>(ws_z, ws_psum, ws_pnsum, out);
}